// IHaarDWT_40870908789396
// MI455X (gfx1250) — compile-verified
//
#include <hip/hip_runtime.h>

#define AS1 __attribute__((address_space(1)))
#define AS3 __attribute__((address_space(3)))

typedef __attribute__((ext_vector_type(2))) float v2f;
typedef __attribute__((ext_vector_type(8))) float v8f;
typedef __attribute__((ext_vector_type(4))) int   v4i;

#define INPUT_LEN 16384
#define NTHREADS  512

#if __has_builtin(__builtin_amdgcn_global_load_async_to_lds_b128)
#define HAVE_ASYNC_LDS 1
#endif
#if __has_builtin(__builtin_amdgcn_wmma_f32_16x16x4_f32)
#define HAVE_WMMA 1
#endif

// ---- constant 16x16 synthesis matrix for the last 4 levels ----
// out[16q + r] = sum_c M[r][c] * v_q[c], with
// v_q = [ z[q], z[q+1024], z[2q+2048+j](j<2), z[4q+4096+j](j<4), z[8q+8192+j](j<8) ]
__device__ __forceinline__ float mval(int r, int c) {
  // c==0 -> s^4 ; group p = floor(log2 c): weight s^(4-p), selected by r>>(4-p)
  if (c == 0) return 0.25f;
  const float mags[4] = {0.25f, 0.35355339059327373f, 0.5f, 0.70710678118654752f};
  int p  = 31 - __clz(c);       // 0..3
  int sh = 4 - p;               // 4..1
  int j  = c - (1 << p);
  if (j != (r >> sh)) return 0.0f;
  float m = mags[p];
  return ((r >> (sh - 1)) & 1) ? -m : m;
}

// LDS index of v_q[c]
__device__ __forceinline__ int vidx(int c, int q) {
  if (c == 0) return q;
  int p = 31 - __clz(c);
  return (q << p) + (1024 << p) + (c - (1 << p));
}

__global__ void __launch_bounds__(NTHREADS)
ihaar_kernel(const float* __restrict__ x, float* __restrict__ out) {
  __shared__ float z[INPUT_LEN];           // 64 KB: one full row

  const int tid = threadIdx.x;
  const long long rowbase = (long long)blockIdx.x * INPUT_LEN;
  const float* xr = x + rowbase;
  float* orow = out + rowbase;

  // ---------------- Phase 1: row -> LDS (async CDNA5 path) --------------
#ifdef HAVE_ASYNC_LDS
  #pragma unroll
  for (int i = tid; i < INPUT_LEN / 4; i += NTHREADS) {
    __builtin_amdgcn_global_load_async_to_lds_b128(
        (AS1 v4i*)(xr + 4 * i), (AS3 v4i*)(&z[4 * i]), 0, 0);
  }
#if __has_builtin(__builtin_amdgcn_s_wait_asynccnt)
  __builtin_amdgcn_s_wait_asynccnt(0);
#else
  asm volatile("s_wait_asynccnt 0" ::: "memory");
#endif
#else
  for (int i = tid; i < INPUT_LEN / 4; i += NTHREADS) {
    ((float4*)z)[i] = ((const float4*)xr)[i];
  }
#endif
  __syncthreads();

  // ---------------- Phase 2: levels l=13..4 (m=2..1024) in LDS ----------
  const float s = 0.70710678118654752f;
  for (int nb = 1; nb <= 512; nb <<= 1) {
    float a = 0.f, d = 0.f;
    const bool act = tid < nb;
    if (act) { a = z[tid]; d = z[tid + nb]; }
    __syncthreads();
    if (act) {
      z[2 * tid]     = (a + d) * s;
      z[2 * tid + 1] = (a - d) * s;
    }
    __syncthreads();
  }

  // ---------------- Phase 3: levels l=3..0 as 16x16x4 WMMAs -------------
  const int lane = tid & 31;
  const int wave = tid >> 5;      // 0..15
  const int col  = lane & 15;     // N / row-of-A index
  const int hi   = lane >> 4;     // 0: K=0,1  1: K=2,3

#ifdef HAVE_WMMA
  // A = M columns, constant across tiles. Layout: lanes0-15 row=lane, v0=K=cb v1=cb+1
  v2f A[4];
  #pragma unroll
  for (int k = 0; k < 4; ++k) {
    const int cb = 4 * k + 2 * hi;
    A[k].x = mval(col, cb);
    A[k].y = mval(col, cb + 1);
  }

  for (int t = wave; t < 64; t += 16) {     // 4 tiles per wave
    const int q = 16 * t + col;
    v2f B[4];
    #pragma unroll
    for (int k = 0; k < 4; ++k) {
      const int cb = 4 * k + 2 * hi;
      B[k].x = z[vidx(cb, q)];
      B[k].y = z[vidx(cb + 1, q)];
    }
    v8f acc = {0.f, 0.f, 0.f, 0.f, 0.f, 0.f, 0.f, 0.f};
    #pragma unroll
    for (int k = 0; k < 4; ++k) {
      acc = __builtin_amdgcn_wmma_f32_16x16x4_f32(
          /*neg_a=*/false, A[k], /*neg_b=*/false, B[k],
          /*c_mod=*/(short)0, acc, /*reuse_a=*/false, /*reuse_b=*/false);
    }
    // D element (M=8*hi+j, N=col) -> out[256t + 16*col + 8*hi + j]
    const int o0 = 256 * t + 16 * col + 8 * hi;
    float4 lo  = make_float4(acc[0], acc[1], acc[2], acc[3]);
    float4 hi4 = make_float4(acc[4], acc[5], acc[6], acc[7]);
    *(float4*)(orow + o0)     = lo;
    *(float4*)(orow + o0 + 4) = hi4;
  }
#else
  // Scalar fallback: 5-term synthesis per output
  for (int t = wave; t < 64; t += 16) {
    const int q = 16 * t + col;
    #pragma unroll
    for (int j = 0; j < 8; ++j) {
      const int r = 8 * hi + j;
      float v = 0.f;
      #pragma unroll
      for (int c = 0; c < 16; ++c) v += mval(r, c) * z[vidx(c, q)];
      orow[16 * q + r] = v;
    }
  }
#endif
}

extern "C" void kernel_launch(void* const* d_in, const int* in_sizes, int n_in,
                              void* d_out, int out_size, void* d_ws, size_t ws_size,
                              hipStream_t stream) {
  const float* x = (const float*)d_in[0];
  float* out = (float*)d_out;
  const int rows = in_sizes[0] / INPUT_LEN;   // 32*128 = 4096
  ihaar_kernel<<<dim3(rows), dim3(NTHREADS), 0, stream>>>(x, out);
}